// BertL2PredictionHead_3384434229585
// MI455X (gfx1250) — compile-verified
//
#include <hip/hip_runtime.h>

typedef __attribute__((ext_vector_type(2))) float v2f;
typedef __attribute__((ext_vector_type(4))) float v4f;
typedef __attribute__((ext_vector_type(8))) float v8f;
typedef __attribute__((ext_vector_type(4))) unsigned int v4u;
typedef __attribute__((ext_vector_type(8))) int v8i;
typedef __attribute__((ext_vector_type(4))) int v4i;

#define V_DIM   20001
#define H_DIM   128
#define M_DIM   2048
#define TILE_N  64
#define ROW_PAD 132   // 128 + 4 floats pad -> ds_load_b64 fragment reads are bank-conflict-free

__global__ __launch_bounds__(256) void l2head_wmma_kernel(
    const float* __restrict__ x,     // (2048, 128)
    const float* __restrict__ emb,   // (20001, 128)
    const float* __restrict__ bias,  // (20001)
    float* __restrict__ out)         // (2048, 20001)
{
    __shared__ float sE[TILE_N * ROW_PAD];
    __shared__ float sEsq[TILE_N];

    const int tid  = threadIdx.x;
    const int lane = tid & 31;
    const int wave = tid >> 5;
    const int r    = lane & 15;   // M-row (A) / N-col (B) within 16
    const int h    = lane >> 4;   // half: selects K pair {0,1} vs {2,3}

    const int n0 = blockIdx.x * TILE_N;
    const int m0 = blockIdx.y * 128 + wave * 16;

#if __has_builtin(__builtin_amdgcn_tensor_load_to_lds)
    // ---- TDM: DMA the 64x128 fp32 E tile into LDS with hardware row padding ----
    // D# group1: data_size=4B, pad every 128 DWORDs by 4 DWORDs -> LDS pitch 132 floats.
    // tensor_dim1 = V - n0 gives hardware OOB zero-fill for the ragged last tile.
    if (wave == 0) {
        const unsigned lds_addr = (unsigned)(size_t)(void*)sE;
        const unsigned long long ga =
            (unsigned long long)(size_t)(const void*)(emb + (size_t)n0 * H_DIM);
        const unsigned rows_left = (unsigned)(V_DIM - n0);   // always > 0
        v4u g0 = { 1u,                                        // count=1 (valid user D#)
                   lds_addr,                                  // lds_addr[31:0]
                   (unsigned)(ga & 0xffffffffu),              // global_addr[31:0]
                   (unsigned)(ga >> 32) | (2u << 30) };       // global_addr[56:32] | type=2
        v8i g1 = { (int)((2u << 16)     // data_size = 4 bytes
                       | (1u << 20)     // pad_enable
                       | (6u << 22)     // pad_interval: every 128 DWORDs
                       | (3u << 25)),   // pad_amount: 4 DWORDs
                   (int)(128u << 16),                              // tensor_dim0=128 (lo16)
                   (int)((rows_left & 0xffffu) << 16),             // dim0 hi | tensor_dim1 lo16
                   (int)(((rows_left >> 16) & 0xffffu)
                       | (128u << 16)),                            // dim1 hi | tile_dim0=128
                   (int)64,                                        // tile_dim1=64, tile_dim2=0
                   (int)128,                                       // tensor_dim0_stride=128 (lo32)
                   (int)(128u << 16),                              // stride0 hi | stride1 lo16
                   0 };                                            // stride1 hi
        v4i gz  = { 0, 0, 0, 0 };                                  // groups 2/3: unused (2D)
        v8i gz8 = { 0, 0, 0, 0, 0, 0, 0, 0 };                      // extra group (clang-23 form)
        __builtin_amdgcn_tensor_load_to_lds(g0, g1, gz, gz, gz8, 0);
    }
#else
    // ---- fallback: manual staging (coalesced 16B loads / stores) ----
    {
        const int row = tid >> 2;     // 0..63
        const int q   = tid & 3;
        const int v   = n0 + row;
        const float* src = emb + (size_t)v * H_DIM;
        float* dst = sE + row * ROW_PAD;
#pragma unroll
        for (int i = 0; i < 8; ++i) {
            const int f4 = i * 4 + q;             // float4 index 0..31
            v4f val = {0.f, 0.f, 0.f, 0.f};
            if (v < V_DIM) val = *(const v4f*)(src + f4 * 4);
            *(v4f*)(dst + f4 * 4) = val;
        }
    }
#endif

    // ---- overlap with the DMA: preload A fragments (16x128 X strip) ----
    // f32 A layout (16x4): lane holds row r; VGPR0 = K=2h, VGPR1 = K=2h+1
    v2f a[32];
    const float* xrow = x + (size_t)(m0 + r) * H_DIM + 2 * h;
#pragma unroll
    for (int kk = 0; kk < 32; ++kk)
        a[kk] = *(const v2f*)(xrow + 4 * kk);

    // per-lane partial ||x||^2; halves hold complementary K pairs
    float part = 0.f;
#pragma unroll
    for (int kk = 0; kk < 32; ++kk)
        part += a[kk].x * a[kk].x + a[kk].y * a[kk].y;
    float xsq = part + __shfl_xor(part, 16, 32);  // lanes r and r+16: ||x[m0+r]||^2

#if __has_builtin(__builtin_amdgcn_tensor_load_to_lds)
    if (wave == 0)
        __builtin_amdgcn_s_wait_tensorcnt(0);     // LDS tile landed
#endif
    __syncthreads();

    // ---- ||e||^2 for the 64 staged rows ----
    if (tid < TILE_N) {
        const float* p = sE + tid * ROW_PAD;
        float s = 0.f;
#pragma unroll 8
        for (int c = 0; c < H_DIM; ++c) { float t = p[c]; s += t * t; }
        sEsq[tid] = s;
    }
    __syncthreads();

    // ---- GEMM: 4 N-tiles, K=128 via 32 x v_wmma_f32_16x16x4_f32 each ----
    v8f acc[4] = {};
#pragma unroll
    for (int kk = 0; kk < 32; ++kk) {
#pragma unroll
        for (int nt = 0; nt < 4; ++nt) {
            const v2f b = *(const v2f*)(sE + (nt * 16 + r) * ROW_PAD + 4 * kk + 2 * h);
            acc[nt] = __builtin_amdgcn_wmma_f32_16x16x4_f32(
                false, a[kk], false, b, (short)0, acc[nt], false, false);
        }
    }

    // ---- epilogue: out = bias - sqrt(max(xsq + esq - 2*dot, 0)) ----
#pragma unroll
    for (int nt = 0; nt < 4; ++nt) {
        const int   n  = n0 + nt * 16 + r;
        const float es = sEsq[nt * 16 + r];
        const bool  ok = (n < V_DIM);
        const float bi = ok ? bias[n] : 0.f;
#pragma unroll
        for (int g = 0; g < 8; ++g) {
            const int   rm = g + 8 * h;              // C layout: VGPR g -> rows g / g+8
            const float xs = __shfl(xsq, rm, 32);
            if (ok) {
                float d2 = xs + es - 2.f * acc[nt][g];
                d2 = fmaxf(d2, 0.f);
                out[(size_t)(m0 + rm) * V_DIM + n] = bi - sqrtf(d2);
            }
        }
    }
}

extern "C" void kernel_launch(void* const* d_in, const int* in_sizes, int n_in,
                              void* d_out, int out_size, void* d_ws, size_t ws_size,
                              hipStream_t stream) {
    (void)in_sizes; (void)n_in; (void)d_ws; (void)ws_size; (void)out_size;
    const float* x    = (const float*)d_in[0];
    const float* emb  = (const float*)d_in[1];
    const float* bias = (const float*)d_in[2];
    float* out = (float*)d_out;

    dim3 grid((V_DIM + TILE_N - 1) / TILE_N, M_DIM / 128, 1);   // 313 x 16
    dim3 block(256, 1, 1);                                      // 8 wave32s
    l2head_wmma_kernel<<<grid, block, 0, stream>>>(x, emb, bias, out);
}